// GATTP_14903536517938
// MI455X (gfx1250) — compile-verified
//
#include <hip/hip_runtime.h>
#include <hip/hip_bf16.h>

typedef __attribute__((ext_vector_type(2))) float v2f;
typedef __attribute__((ext_vector_type(8))) float v8f;

#define NUM_GRAPHS 512
#define NN      100000
#define DIN     1024
#define DD      64
#define HH      32
#define KT      64        // K-chunk of W_enc staged in LDS
#define WSTR    68        // padded LDS row stride (68 % 64 == 4 -> conflict-free b64)
#define XSTR    68        // padded LDS row stride for xe staging

// ---------------------------------------------------------------------------
// Kernel A: xe = x @ W_enc^T + b_enc   (f32 WMMA 16x16x4)
//           gate = xe @ W_gate^T + b_gate (f32 WMMA 16x16x4, xe via LDS)
// 256 threads = 8 waves, each wave owns one 16-row tile (128 rows / block).
// ---------------------------------------------------------------------------
__global__ __launch_bounds__(256) void enc_gate_kernel(
    const float* __restrict__ x, const float* __restrict__ W_enc,
    const float* __restrict__ b_enc, const float* __restrict__ W_gate,
    const float* __restrict__ b_gate, float* __restrict__ xe,
    float* __restrict__ gate)
{
  __shared__ float Ws[DD * WSTR];          // 64 x 68 f32  (~17.4 KB)
  __shared__ float Xe[8 * 16 * XSTR];      // 8 waves x 16 x 68 (~34.8 KB)

  const int tid  = threadIdx.x;
  const int w    = tid >> 5;
  const int lane = tid & 31;
  const int half = lane >> 4;              // 0: K pair {0,1}, 1: K pair {2,3}
  const int l16  = lane & 15;
  const int row0 = (blockIdx.x * 8 + w) * 16;
  const bool active = (row0 < NN);

  v8f acc0 = {}, acc1 = {}, acc2 = {}, acc3 = {};

  for (int kt = 0; kt < DIN; kt += KT) {
    // cooperative coalesced stage of W_enc[:, kt:kt+KT] -> LDS
    for (int i = tid; i < DD * KT; i += 256) {
      int n = i >> 6;
      int k = i & 63;
      Ws[n * WSTR + k] = W_enc[n * DIN + kt + k];
    }
    __syncthreads();

    if (active) {
      const float* xrow = x + (size_t)(row0 + l16) * DIN + kt + half * 2;
      #pragma unroll 4
      for (int kc = 0; kc < KT; kc += 4) {
        v2f a = *(const v2f*)(xrow + kc);                     // global_load_b64
        v2f b0 = *(const v2f*)(&Ws[( 0 + l16) * WSTR + kc + half * 2]);
        acc0 = __builtin_amdgcn_wmma_f32_16x16x4_f32(false, a, false, b0,
                                                     (short)0, acc0, false, false);
        v2f b1 = *(const v2f*)(&Ws[(16 + l16) * WSTR + kc + half * 2]);
        acc1 = __builtin_amdgcn_wmma_f32_16x16x4_f32(false, a, false, b1,
                                                     (short)0, acc1, false, false);
        v2f b2 = *(const v2f*)(&Ws[(32 + l16) * WSTR + kc + half * 2]);
        acc2 = __builtin_amdgcn_wmma_f32_16x16x4_f32(false, a, false, b2,
                                                     (short)0, acc2, false, false);
        v2f b3 = *(const v2f*)(&Ws[(48 + l16) * WSTR + kc + half * 2]);
        acc3 = __builtin_amdgcn_wmma_f32_16x16x4_f32(false, a, false, b3,
                                                     (short)0, acc3, false, false);
      }
    }
    __syncthreads();
  }

  // epilogue: bias, write xe, stage tile to wave-private LDS for the gate GEMM
  float* xew = &Xe[w * 16 * XSTR];
  if (active) {
    #pragma unroll
    for (int r = 0; r < 8; r++) {
      int lrow = r + half * 8;
      float v;
      v = acc0[r] + b_enc[ 0 + l16];
      xe[(size_t)(row0 + lrow) * DD +  0 + l16] = v;  xew[lrow * XSTR +  0 + l16] = v;
      v = acc1[r] + b_enc[16 + l16];
      xe[(size_t)(row0 + lrow) * DD + 16 + l16] = v;  xew[lrow * XSTR + 16 + l16] = v;
      v = acc2[r] + b_enc[32 + l16];
      xe[(size_t)(row0 + lrow) * DD + 32 + l16] = v;  xew[lrow * XSTR + 32 + l16] = v;
      v = acc3[r] + b_enc[48 + l16];
      xe[(size_t)(row0 + lrow) * DD + 48 + l16] = v;  xew[lrow * XSTR + 48 + l16] = v;
    }
  }
  __syncthreads();   // block-uniform; orders LDS stores vs. loads below

  if (active) {
    v8f g0 = {}, g1 = {};
    #pragma unroll 4
    for (int kc = 0; kc < DD; kc += 4) {
      v2f a = *(const v2f*)(&xew[l16 * XSTR + kc + half * 2]);      // ds_load_b64
      v2f b0 = *(const v2f*)(W_gate + (size_t)( 0 + l16) * DD + kc + half * 2);
      g0 = __builtin_amdgcn_wmma_f32_16x16x4_f32(false, a, false, b0,
                                                 (short)0, g0, false, false);
      v2f b1 = *(const v2f*)(W_gate + (size_t)(16 + l16) * DD + kc + half * 2);
      g1 = __builtin_amdgcn_wmma_f32_16x16x4_f32(false, a, false, b1,
                                                 (short)0, g1, false, false);
    }
    #pragma unroll
    for (int r = 0; r < 8; r++) {
      int lrow = r + half * 8;
      gate[(size_t)(row0 + lrow) * HH +  0 + l16] = g0[r] + b_gate[ 0 + l16];
      gate[(size_t)(row0 + lrow) * HH + 16 + l16] = g1[r] + b_gate[16 + l16];
    }
  }
}

// ---------------------------------------------------------------------------
// Kernel B: per-graph (segment) softmax over gate + weighted pool + ReLU.
// One block per graph. batch is sorted -> binary search segment bounds.
// ---------------------------------------------------------------------------
__global__ __launch_bounds__(256) void pool_kernel(
    const int* __restrict__ batch, const float* __restrict__ xe,
    const float* __restrict__ gate, float* __restrict__ out)
{
  __shared__ float smax[8 * HH];
  __shared__ float ssum[8 * HH];
  __shared__ float gmax_s[HH];
  __shared__ float gscale_s[HH];

  const int b    = blockIdx.x;
  const int tid  = threadIdx.x;
  const int w    = tid >> 5;
  const int lane = tid & 31;

  // segment bounds via binary search (batch sorted ascending)
  int lo = 0, hi = NN;
  while (lo < hi) { int mid = (lo + hi) >> 1; if (batch[mid] < b) lo = mid + 1; else hi = mid; }
  const int start = lo;
  lo = start; hi = NN;
  while (lo < hi) { int mid = (lo + hi) >> 1; if (batch[mid] < b + 1) lo = mid + 1; else hi = mid; }
  const int end = lo;

  // pass 1: per-head max (lane = head, waves stride nodes)
  float m = -3.402823466e38f;
  for (int n = start + w; n < end; n += 8) m = fmaxf(m, gate[(size_t)n * HH + lane]);
  smax[w * HH + lane] = m;
  __syncthreads();
  if (w == 0) {
    float mm = smax[lane];
    #pragma unroll
    for (int i = 1; i < 8; i++) mm = fmaxf(mm, smax[i * HH + lane]);
    gmax_s[lane] = mm;
  }
  __syncthreads();

  // pass 2: per-head sum of exp
  const float gm = gmax_s[lane];
  float s = 0.f;
  for (int n = start + w; n < end; n += 8) s += __expf(gate[(size_t)n * HH + lane] - gm);
  ssum[w * HH + lane] = s;
  __syncthreads();
  if (w == 0) {
    float ss = ssum[lane];
    #pragma unroll
    for (int i = 1; i < 8; i++) ss += ssum[i * HH + lane];
    gscale_s[lane] = (ss > 0.f) ? (1.0f / ss) : 0.f;
  }
  __syncthreads();

  // pass 3: wave w pools heads 4w..4w+3; lane covers d = lane and lane+32
  const int h0 = w * 4;
  const float gm0 = gmax_s[h0 + 0], gm1 = gmax_s[h0 + 1];
  const float gm2 = gmax_s[h0 + 2], gm3 = gmax_s[h0 + 3];
  float a00 = 0.f, a01 = 0.f, a02 = 0.f, a03 = 0.f;
  float a10 = 0.f, a11 = 0.f, a12 = 0.f, a13 = 0.f;
  for (int n = start; n < end; n++) {
    const float xl = xe[(size_t)n * DD + lane];
    const float xh = xe[(size_t)n * DD + lane + 32];
    const float* gp = gate + (size_t)n * HH + h0;
    const float w0 = __expf(gp[0] - gm0);
    const float w1 = __expf(gp[1] - gm1);
    const float w2 = __expf(gp[2] - gm2);
    const float w3 = __expf(gp[3] - gm3);
    a00 = fmaf(w0, xl, a00); a10 = fmaf(w0, xh, a10);
    a01 = fmaf(w1, xl, a01); a11 = fmaf(w1, xh, a11);
    a02 = fmaf(w2, xl, a02); a12 = fmaf(w2, xh, a12);
    a03 = fmaf(w3, xl, a03); a13 = fmaf(w3, xh, a13);
  }
  const float s0 = gscale_s[h0 + 0], s1 = gscale_s[h0 + 1];
  const float s2 = gscale_s[h0 + 2], s3 = gscale_s[h0 + 3];
  float* ob = out + (size_t)b * (HH * DD);
  ob[(h0 + 0) * DD + lane]      = fmaxf(a00 * s0, 0.f);
  ob[(h0 + 0) * DD + lane + 32] = fmaxf(a10 * s0, 0.f);
  ob[(h0 + 1) * DD + lane]      = fmaxf(a01 * s1, 0.f);
  ob[(h0 + 1) * DD + lane + 32] = fmaxf(a11 * s1, 0.f);
  ob[(h0 + 2) * DD + lane]      = fmaxf(a02 * s2, 0.f);
  ob[(h0 + 2) * DD + lane + 32] = fmaxf(a12 * s2, 0.f);
  ob[(h0 + 3) * DD + lane]      = fmaxf(a03 * s3, 0.f);
  ob[(h0 + 3) * DD + lane + 32] = fmaxf(a13 * s3, 0.f);
}

// ---------------------------------------------------------------------------
extern "C" void kernel_launch(void* const* d_in, const int* in_sizes, int n_in,
                              void* d_out, int out_size, void* d_ws, size_t ws_size,
                              hipStream_t stream) {
  const float* x      = (const float*)d_in[0];
  const int*   batch  = (const int*)  d_in[1];
  const float* W_enc  = (const float*)d_in[2];
  const float* b_enc  = (const float*)d_in[3];
  const float* W_gate = (const float*)d_in[4];
  const float* b_gate = (const float*)d_in[5];
  float* out = (float*)d_out;

  float* xe   = (float*)d_ws;                  // [N, 64]
  float* gate = xe + (size_t)NN * DD;          // [N, 32]

  const int rowTiles = (NN + 15) / 16;         // 6250 (exact)
  const int blocksA  = (rowTiles + 7) / 8;     // 782

  enc_gate_kernel<<<blocksA, 256, 0, stream>>>(x, W_enc, b_enc, W_gate, b_gate,
                                               xe, gate);
  pool_kernel<<<NUM_GRAPHS, 256, 0, stream>>>(batch, xe, gate, out);
}